// HyperbolicMLR_31078383354263
// MI455X (gfx1250) — compile-verified
//
#include <hip/hip_runtime.h>
#include <math.h>

// ---------------------------------------------------------------------------
// HyperbolicMLR for MI455X (gfx1250, wave32).
//
// logits[b,c] = scale_c * asinh( numer / denom ) where the Mobius-add terms
// factor into two D=64 contractions:
//   XY[b,c] = dot(x_b, mp_c)        (mp = -p_poin)
//   XA[b,c] = dot(x_b, a_poin_c)
// computed with V_WMMA_F32_16X16X4_F32 (full f32 precision; problem is
// HBM-store bound at ~9MB total traffic, so f32 tensor ops are free).
// Epilogue uses native TRANS ops (v_rcp/v_sqrt/v_log) so the asinh tail
// stays under the memory roofline.
// ---------------------------------------------------------------------------

typedef __attribute__((ext_vector_type(2))) float v2f;
typedef __attribute__((ext_vector_type(8))) float v8f;

#define HMLR_K    0.1f
#define HMLR_SK   0.31622776601683794f   /* sqrt(0.1) */
#define HMLR_EPS  1e-15f
#define HMLR_LN2  0.6931471805599453f

// ---------------------------------------------------------------------------
// Phase 1a: per-class preprocessing. One wave (32 lanes) per class, D=64 so
// each lane owns elements {lane, lane+32}. Butterfly reductions via shfl_xor.
// Outputs: mp[c][64], a_poin[c][64], cscal[c] = {x2, dot(mp,a_poin), a_norm, scale}
// ---------------------------------------------------------------------------
__global__ void __launch_bounds__(256)
hmlr_prep_classes(const float* __restrict__ a_vals,
                  const float* __restrict__ p_vals,
                  float* __restrict__ mp,
                  float* __restrict__ ap,
                  float4* __restrict__ cscal,
                  int C) {
    int wave = (blockIdx.x * blockDim.x + threadIdx.x) >> 5;
    int lane = threadIdx.x & 31;
    if (wave >= C) return;

    const float k = HMLR_K, sk = HMLR_SK;
    size_t base = (size_t)wave * 64 + lane;
    float p0 = p_vals[base], p1 = p_vals[base + 32];
    float a0 = a_vals[base], a1 = a_vals[base + 32];

    // ||p||^2 across the wave
    float pn2 = p0 * p0 + p1 * p1;
    #pragma unroll
    for (int off = 16; off; off >>= 1) pn2 += __shfl_xor(pn2, off, 32);

    float pnorm = fmaxf(sqrtf(pn2), HMLR_EPS);
    float t = tanhf(sk * pnorm) / (sk * pnorm);     // p_poin = t * p
    float pp0 = t * p0, pp1 = t * p1;
    float p2 = t * t * pn2;                         // ||p_poin||^2 == x2

    float ka = 1.0f + k * p2;
    float ap0 = a0 * ka, ap1 = a1 * ka;             // a_poin

    float a2 = ap0 * ap0 + ap1 * ap1;
    float pa = -(pp0 * ap0 + pp1 * ap1);            // dot(mp, a_poin)
    #pragma unroll
    for (int off = 16; off; off >>= 1) {
        a2 += __shfl_xor(a2, off, 32);
        pa += __shfl_xor(pa, off, 32);
    }

    float anorm = fmaxf(sqrtf(a2), HMLR_EPS);
    float lam   = 2.0f / (1.0f - k * p2);
    float scale = lam * anorm / sk;

    mp[base]      = -pp0;
    mp[base + 32] = -pp1;
    ap[base]      = ap0;
    ap[base + 32] = ap1;
    if (lane == 0) cscal[wave] = make_float4(p2, pa, anorm, scale);
}

// ---------------------------------------------------------------------------
// Phase 1b: y2[b] = ||x_b||^2. One wave per batch row.
// ---------------------------------------------------------------------------
__global__ void __launch_bounds__(256)
hmlr_prep_y2(const float* __restrict__ x, float* __restrict__ y2, int B) {
    int wave = (blockIdx.x * blockDim.x + threadIdx.x) >> 5;
    int lane = threadIdx.x & 31;
    if (wave >= B) return;
    size_t base = (size_t)wave * 64 + lane;
    float v0 = x[base], v1 = x[base + 32];
    float s = v0 * v0 + v1 * v1;
    #pragma unroll
    for (int off = 16; off; off >>= 1) s += __shfl_xor(s, off, 32);
    if (lane == 0) y2[wave] = s;
}

// ---------------------------------------------------------------------------
// Phase 2: main kernel. One wave per 16x16 output tile (16 batch rows x 16
// classes). K=64 contraction done as 16 x V_WMMA_F32_16X16X4_F32 per matrix.
//
// Fragment addressing (from ISA 7.12.2, 32-bit 16x4 A / 4x16 B layouts):
//   A (X tile):  lane holds row m=lane&15, K pair {2*(lane>>4), +1} -> float2
//   B (mp/ap^T): lane holds col n=lane&15, K pair {2*(lane>>4), +1} -> float2
//   D:           vgpr v, lane -> row = 16*tileB + v + 8*(lane>>4), col = lane&15
// ---------------------------------------------------------------------------
__global__ void __launch_bounds__(256)
hmlr_main(const float* __restrict__ x,
          const float* __restrict__ mp,
          const float* __restrict__ ap,
          const float4* __restrict__ cscal,
          const float* __restrict__ y2,
          float* __restrict__ out,
          int B, int C, int nTilesC, int nTiles) {
    int wave = threadIdx.x >> 5;
    int lane = threadIdx.x & 31;
    int tile = blockIdx.x * (blockDim.x >> 5) + wave;
    if (tile >= nTiles) return;                    // wave-uniform: EXEC stays all-1s

    int tileB = tile / nTilesC;
    int tileC = tile - tileB * nTilesC;

    int m  = lane & 15;
    int kh = lane >> 4;                            // K half-select (0 or 1)
    int bRow = tileB * 16 + m;
    int cG   = tileC * 16 + m;                     // global class of this lane
    int cIdx = (cG < C) ? cG : (C - 1);            // clamp padded classes

    const float* xr = x  + (size_t)bRow * 64 + 2 * kh;
    const float* mr = mp + (size_t)cIdx * 64 + 2 * kh;
    const float* ar = ap + (size_t)cIdx * 64 + 2 * kh;

    v8f axy = {0.f, 0.f, 0.f, 0.f, 0.f, 0.f, 0.f, 0.f};
    v8f axa = {0.f, 0.f, 0.f, 0.f, 0.f, 0.f, 0.f, 0.f};

    #pragma unroll
    for (int s = 0; s < 16; ++s) {
        v2f af = *(const v2f*)(xr + 4 * s);
        v2f bm = *(const v2f*)(mr + 4 * s);
        v2f ba = *(const v2f*)(ar + 4 * s);
        axy = __builtin_amdgcn_wmma_f32_16x16x4_f32(
            false, af, false, bm, (short)0, axy, false, false);
        axa = __builtin_amdgcn_wmma_f32_16x16x4_f32(
            false, af, false, ba, (short)0, axa, false, false);
    }

    // epilogue: per-class scalars (uniform over the 8 accumulator rows)
    float4 cs = cscal[cIdx];
    float x2 = cs.x, pa = cs.y, anorm = cs.z, scale = cs.w;
    const float k = HMLR_K, sk = HMLR_SK;
    int rbase = tileB * 16 + 8 * kh;

    #pragma unroll
    for (int v = 0; v < 8; ++v) {
        int row = rbase + v;
        float yy = y2[row];
        float xy = axy[v];
        float xa = axa[v];

        float alpha = fmaf(2.0f * k, xy, fmaf(k, yy, 1.0f));
        float beta  = fmaf(-k, x2, 1.0f);
        float den   = fmaf(2.0f * k, xy, fmaf(k * k * x2, yy, 1.0f));
        float id    = __builtin_amdgcn_rcpf(den);          // v_rcp_f32

        float mda  = (alpha * pa + beta * xa) * id;        // dot(mob, a_poin)
        float mob2 = (alpha * alpha * x2 + 2.0f * alpha * beta * xy
                      + beta * beta * yy) * (id * id);     // ||mob||^2

        float numer = 2.0f * sk * mda;
        float denom = anorm * fmaf(-k, mob2, 1.0f);
        float z  = numer * __builtin_amdgcn_rcpf(denom);   // v_rcp_f32

        // asinh(z) = sign(z) * ln2 * log2(|z| + sqrt(z^2 + 1)) on TRANS units
        float az = fabsf(z);
        float sq = __builtin_amdgcn_sqrtf(fmaf(az, az, 1.0f));  // v_sqrt_f32
        float lg = __builtin_amdgcn_logf(az + sq);              // v_log_f32 (log2)
        float r  = scale * (HMLR_LN2 * lg);
        float logit = (z < 0.0f) ? -r : r;

        if (cG < C) out[(size_t)row * C + cG] = logit;
    }
}

// ---------------------------------------------------------------------------
// Host launcher. Inputs: d_in[0]=x [B,64], d_in[1]=a_vals [C,64],
// d_in[2]=p_vals [C,64]; output: logits [B,C] f32.
// Workspace layout: mp (C*64 f32) | ap (C*64 f32) | cscal (C float4) | y2 (B f32)
// ---------------------------------------------------------------------------
extern "C" void kernel_launch(void* const* d_in, const int* in_sizes, int n_in,
                              void* d_out, int out_size, void* d_ws, size_t ws_size,
                              hipStream_t stream) {
    const float* x      = (const float*)d_in[0];
    const float* a_vals = (const float*)d_in[1];
    const float* p_vals = (const float*)d_in[2];
    float* out = (float*)d_out;

    const int D = 64;
    int B = in_sizes[0] / D;   // 2048
    int C = in_sizes[1] / D;   // 1000

    char* ws = (char*)d_ws;
    size_t rowBytes = (size_t)C * D * sizeof(float);     // multiple of 256B
    float*  mp = (float*)(ws);
    float*  ap = (float*)(ws + rowBytes);
    float4* cs = (float4*)(ws + 2 * rowBytes);
    float*  y2 = (float*)(ws + 2 * rowBytes + (size_t)C * sizeof(float4));

    // 8 waves per 256-thread block throughout
    hmlr_prep_classes<<<(C + 7) / 8, 256, 0, stream>>>(a_vals, p_vals, mp, ap, cs, C);
    hmlr_prep_y2<<<(B + 7) / 8, 256, 0, stream>>>(x, y2, B);

    int nTilesB = B / 16;               // 128 (B divisible by 16)
    int nTilesC = (C + 15) / 16;        // 63
    int nTiles  = nTilesB * nTilesC;    // 8064
    hmlr_main<<<(nTiles + 7) / 8, 256, 0, stream>>>(x, mp, ap, cs, y2, out,
                                                    B, C, nTilesC, nTiles);
}